// AssignIndex_21844203667947
// MI455X (gfx1250) — compile-verified
//
#include <hip/hip_runtime.h>

// ---------------------------------------------------------------------------
// AssignIndex: out = arr (65536 x 1024 fp32) with out[index, :] = element.
// Pure streaming op: ~0 FLOPs, 0.54 GB of traffic -> HBM-bound (~23 us floor
// at 23.3 TB/s). WMMA is irrelevant (arithmetic intensity ~0); the CDNA5
// feature that matters is the data-movement path: B128 accesses and the
// async Global<->LDS DMA engine (ASYNCcnt-tracked) so bulk data never
// occupies VGPRs. Row fixup is a separate tiny kernel, stream-ordered after
// the bulk copy.
// ---------------------------------------------------------------------------

#define BLOCK 256
#define STAGES 4          // 4 x (256 threads x 16B) = 16 KiB LDS per block

typedef float f4  __attribute__((ext_vector_type(4)));
typedef int   v4i __attribute__((ext_vector_type(4)));
typedef __attribute__((address_space(1))) v4i* gv4i_p;   // global (AS1)
typedef __attribute__((address_space(3))) v4i* lv4i_p;   // LDS    (AS3)

#if __has_builtin(__builtin_amdgcn_global_load_async_to_lds_b128) && \
    __has_builtin(__builtin_amdgcn_global_store_async_from_lds_b128)
#define HAVE_ASYNC_LDS 1
#else
#define HAVE_ASYNC_LDS 0
#endif

#if __has_builtin(__builtin_amdgcn_s_wait_asynccnt)
#define WAIT_ASYNC0() __builtin_amdgcn_s_wait_asynccnt(0)
#else
#define WAIT_ASYNC0() asm volatile("s_wait_asynccnt 0" ::: "memory")
#endif

#if HAVE_ASYNC_LDS
// Bulk copy via CDNA5 async Global->LDS->Global DMA. Per-thread-disjoint LDS
// slots mean the only synchronization needed is s_wait_asynccnt between the
// load-issue and store-issue phases (async ops of one wave are ordered by
// ASYNCcnt; no cross-thread LDS sharing -> no block barrier).
__global__ void copy_async_kernel(const float* __restrict__ src,
                                  float* __restrict__ dst,
                                  long total_f4, int iters_per_block) {
  __shared__ __align__(16) v4i lds_buf[BLOCK * STAGES];
  lv4i_p lds   = (lv4i_p)lds_buf;
  gv4i_p g_src = (gv4i_p)const_cast<float*>(src);
  gv4i_p g_dst = (gv4i_p)dst;

  const int tid = threadIdx.x;
  const long f4_per_iter = (long)BLOCK * STAGES;
  const long base_f4 = (long)blockIdx.x * iters_per_block * f4_per_iter;

  for (int it = 0; it < iters_per_block; ++it) {
    const long iter_f4 = base_f4 + (long)it * f4_per_iter;
#pragma unroll
    for (int s = 0; s < STAGES; ++s) {
      const long gi = iter_f4 + (long)s * BLOCK + tid;
      if (gi < total_f4) {
        __builtin_amdgcn_global_load_async_to_lds_b128(
            g_src + gi, lds + (s * BLOCK + tid), /*offset=*/0, /*cpol=*/0);
      }
    }
    WAIT_ASYNC0();   // loads landed in LDS
#pragma unroll
    for (int s = 0; s < STAGES; ++s) {
      const long gi = iter_f4 + (long)s * BLOCK + tid;
      if (gi < total_f4) {
        __builtin_amdgcn_global_store_async_from_lds_b128(
            g_dst + gi, lds + (s * BLOCK + tid), /*offset=*/0, /*cpol=*/0);
      }
    }
    WAIT_ASYNC0();   // LDS slots free for reuse next iteration
  }
}
#endif  // HAVE_ASYNC_LDS

// Fallback / reference-quality path: grid-stride B128 copy with non-temporal
// cache policy on both sides (global_load_b128 / global_store_b128, TH=NT).
__global__ void copy_nt_kernel(const f4* __restrict__ src,
                               f4* __restrict__ dst, long total_f4) {
  const long stride = (long)gridDim.x * blockDim.x;
  long gi = (long)blockIdx.x * blockDim.x + threadIdx.x;
#pragma unroll 4
  for (; gi < total_f4; gi += stride) {
    f4 v = __builtin_nontemporal_load(&src[gi]);
    __builtin_nontemporal_store(v, &dst[gi]);
  }
}

// Scatter element into row *index_p. Runs after the bulk copy on the same
// stream, so it deterministically overwrites whatever the copy wrote there.
__global__ void fixup_row_kernel(const int* __restrict__ index_p,
                                 const float* __restrict__ element,
                                 float* __restrict__ out, int d) {
  const int row = index_p[0];
  const int i4 = blockIdx.x * blockDim.x + threadIdx.x;
  const int d4 = d >> 2;
  if (i4 < d4) {
    const f4* e4 = (const f4*)element;
    f4* o4 = (f4*)out + (long)row * d4;
    o4[i4] = e4[i4];
  }
}

extern "C" void kernel_launch(void* const* d_in, const int* in_sizes, int n_in,
                              void* d_out, int out_size, void* d_ws, size_t ws_size,
                              hipStream_t stream) {
  const float* arr  = (const float*)d_in[0];
  const int*   idx  = (const int*)d_in[1];
  const float* elem = (const float*)d_in[2];
  float*       out  = (float*)d_out;

  const long total    = (long)out_size;   // N * D = 65536 * 1024
  const long total_f4 = total >> 2;       // 16,777,216 float4s
  const int  d        = in_sizes[2];      // 1024

#if HAVE_ASYNC_LDS
  const int  iters_per_block = 4;         // 64 KiB copied per block
  const long f4_per_block = (long)BLOCK * STAGES * iters_per_block;
  const long nblocks = (total_f4 + f4_per_block - 1) / f4_per_block;  // 4096
  copy_async_kernel<<<dim3((unsigned)nblocks), dim3(BLOCK), 0, stream>>>(
      arr, out, total_f4, iters_per_block);
#else
  const long f4_per_thread = 8;
  long grid = (total_f4 + (long)BLOCK * f4_per_thread - 1) /
              ((long)BLOCK * f4_per_thread);                           // 8192
  if (grid < 1) grid = 1;
  copy_nt_kernel<<<dim3((unsigned)grid), dim3(BLOCK), 0, stream>>>(
      (const f4*)arr, (f4*)out, total_f4);
#endif

  const int d4 = d >> 2;
  fixup_row_kernel<<<dim3((unsigned)((d4 + BLOCK - 1) / BLOCK)),
                     dim3(BLOCK), 0, stream>>>(idx, elem, out, d);
}